// YOLOLossV1_43748536877617
// MI455X (gfx1250) — compile-verified
//
#include <hip/hip_runtime.h>
#include <hip/hip_bf16.h>
#include <limits.h>

namespace {

constexpr int   kS       = 7;
constexpr int   kBatch   = 16384;
constexpr int   kCells   = kBatch * kS * kS;   // 802816
constexpr int   kD       = 30;                 // 5*B + C = 10 + 20
constexpr int   kThreads = 256;                // 8 wave32 per block
constexpr int   kBlocks  = kCells / kThreads;  // 3136 (exact)
constexpr float kLCoord  = 5.0f;
constexpr float kLNoobj  = 0.5f;

typedef __attribute__((ext_vector_type(2))) float v2f;
typedef __attribute__((ext_vector_type(8))) float v8f;

// Candidate pair for "two smallest flat box indices" (sentinel = INT_MAX).
struct P2 { int f0; float d0; int f1; float d1; };          // f0 <= f1 invariant
struct BlockRec { float sum; int f0; float d0; int f1; float d1; int pad; };

__device__ __forceinline__ P2 merge_p2(P2 a, P2 b) {
    P2 r;
    if (a.f0 <= b.f0) {
        r.f0 = a.f0; r.d0 = a.d0;
        if (a.f1 <= b.f0) { r.f1 = a.f1; r.d1 = a.d1; }
        else              { r.f1 = b.f0; r.d1 = b.d0; }
    } else {
        r.f0 = b.f0; r.d0 = b.d0;
        if (b.f1 <= a.f0) { r.f1 = b.f1; r.d1 = b.d1; }
        else              { r.f1 = a.f0; r.d1 = a.d0; }
    }
    return r;
}

// Wave32 sum reduction on the XDL: D = A(16x4) * ones(4x16).
// A vgpr0 = per-lane value, vgpr1 = 0  ->  D[m][n] = v(lane m) + v(lane m+16).
// Lane j<16 holds column j rows 0..7 in d[0..7]; lane j+16 holds rows 8..15.
// So sum(d[0..7]) + xor16-partner = full 32-lane sum in every lane.
__device__ __forceinline__ float wave_reduce_sum_wmma(float v) {
    v2f a; a[0] = v;    a[1] = 0.0f;
    v2f b; b[0] = 1.0f; b[1] = 1.0f;           // ones in all K rows of B
    v8f c = {};
    v8f d = __builtin_amdgcn_wmma_f32_16x16x4_f32(
        /*neg_a=*/false, a, /*neg_b=*/false, b,
        /*c_mod=*/(short)0, c, /*reuse_a=*/false, /*reuse_b=*/false);
    float s = d[0] + d[1] + d[2] + d[3] + d[4] + d[5] + d[6] + d[7];
    s += __shfl_xor(s, 16, 32);
    return s;
}

__device__ __forceinline__ P2 wave_reduce_p2(P2 p) {
#pragma unroll
    for (int m = 16; m; m >>= 1) {
        P2 o;
        o.f0 = __shfl_xor(p.f0, m, 32);
        o.d0 = __shfl_xor(p.d0, m, 32);
        o.f1 = __shfl_xor(p.f1, m, 32);
        o.d1 = __shfl_xor(p.d1, m, 32);
        p = merge_p2(p, o);
    }
    return p;
}

// Mirrors reference: corners from (x/7 +- 0.5w), areas from corner diffs.
__device__ __forceinline__ float iou_box(float px, float py, float pw, float ph,
                                         float gx0, float gy0, float gx1, float gy1,
                                         float garea) {
    float ax0 = px / 7.0f - 0.5f * pw;
    float ay0 = py / 7.0f - 0.5f * ph;
    float ax1 = px / 7.0f + 0.5f * pw;
    float ay1 = py / 7.0f + 0.5f * ph;
    float iw = fmaxf(fminf(ax1, gx1) - fmaxf(ax0, gx0), 0.0f);
    float ih = fmaxf(fminf(ay1, gy1) - fmaxf(ay0, gy0), 0.0f);
    float inter = iw * ih;
    float aa = (ax1 - ax0) * (ay1 - ay0);
    return inter / (aa + garea - inter);
}

__global__ __launch_bounds__(kThreads)
void yolo_pass1(const float* __restrict__ pred, const float* __restrict__ targ,
                BlockRec* __restrict__ recs) {
    const int tid  = threadIdx.x;
    const int cell = blockIdx.x * kThreads + tid;    // grid covers kCells exactly
    const float* p = pred + (long long)cell * kD;
    const float* t = targ + (long long)cell * kD;

    // Records are 120 B => 8 B aligned; stream with b64 loads.
    v2f pc[5], tc[5];
#pragma unroll
    for (int i = 0; i < 5; ++i) {
        pc[i] = *(const v2f*)(p + 2 * i);
        tc[i] = *(const v2f*)(t + 2 * i);
    }
    float clssum = 0.0f;
#pragma unroll
    for (int i = 0; i < 10; ++i) {
        v2f pv = *(const v2f*)(p + 10 + 2 * i);
        v2f tv = *(const v2f*)(t + 10 + 2 * i);
        float e0 = pv[0] - tv[0], e1 = pv[1] - tv[1];
        clssum += e0 * e0 + e1 * e1;
    }

    const float objf = (tc[0][0] == 1.0f) ? 1.0f : 0.0f;

    // gt box 0 corners (IoU target for both pred boxes)
    float gx0 = tc[1][0] / 7.0f - 0.5f * tc[2][0];
    float gy0 = tc[1][1] / 7.0f - 0.5f * tc[2][1];
    float gx1 = tc[1][0] / 7.0f + 0.5f * tc[2][0];
    float gy1 = tc[1][1] / 7.0f + 0.5f * tc[2][1];
    float garea = (gx1 - gx0) * (gy1 - gy0);

    float iou0 = iou_box(pc[1][0], pc[1][1], pc[2][0], pc[2][1], gx0, gy0, gx1, gy1, garea);
    float iou1 = iou_box(pc[3][0], pc[3][1], pc[4][0], pc[4][1], gx0, gy0, gx1, gy1, garea);

    const bool  r1    = (iou1 > iou0);             // argmax, first-max on ties
    const float iou_r = r1 ? iou1 : iou0;
    const float pcf_r = r1 ? pc[0][1] : pc[0][0];

    // responsible pred/gt box raw values
    float pbx = r1 ? pc[3][0] : pc[1][0], pby = r1 ? pc[3][1] : pc[1][1];
    float pbw = r1 ? pc[4][0] : pc[2][0], pbh = r1 ? pc[4][1] : pc[2][1];
    float gbx = r1 ? tc[3][0] : tc[1][0], gby = r1 ? tc[3][1] : tc[1][1];
    float gbw = r1 ? tc[4][0] : tc[2][0], gbh = r1 ? tc[4][1] : tc[2][1];

    float ex = pbx - gbx, ey = pby - gby, ew = pbw - gbw, eh = pbh - gbh;
    float plain = ex * ex + ey * ey + ew * ew + eh * eh;
    float sx = sqrtf(pbx) - sqrtf(gbx), sy = sqrtf(pby) - sqrtf(gby);
    float sw = sqrtf(pbw) - sqrtf(gbw), sh = sqrtf(pbh) - sqrtf(gbh);
    float sqd = sx * sx + sy * sy + sw * sw + sh * sh;

    float hit   = (pcf_r - iou_r) * (pcf_r - iou_r);
    float noobj = pc[0][0] * pc[0][0] + pc[0][1] * pc[0][1] - objf * pcf_r * pcf_r;
    float contrib = objf * (hit + clssum + kLCoord * sqd) + kLNoobj * noobj;

    P2 mine;
    mine.f0 = (objf != 0.0f) ? (2 * cell + (r1 ? 1 : 0)) : INT_MAX;
    mine.d0 = plain - sqd;
    mine.f1 = INT_MAX; mine.d1 = 0.0f;

    // Branchless above => EXEC is all-ones at the WMMA.
    float ws = wave_reduce_sum_wmma(contrib);
    P2    wp = wave_reduce_p2(mine);

    __shared__ float swsum[kThreads / 32];
    __shared__ P2    swp2[kThreads / 32];
    const int wid = tid >> 5, lane = tid & 31;
    if (lane == 0) { swsum[wid] = ws; swp2[wid] = wp; }
    __syncthreads();
    if (tid == 0) {
        float bs = 0.0f;
        P2 acc = swp2[0];
#pragma unroll
        for (int i = 0; i < kThreads / 32; ++i) bs += swsum[i];
#pragma unroll
        for (int i = 1; i < kThreads / 32; ++i) acc = merge_p2(acc, swp2[i]);
        BlockRec r; r.sum = bs; r.f0 = acc.f0; r.d0 = acc.d0;
        r.f1 = acc.f1; r.d1 = acc.d1; r.pad = 0;
        recs[blockIdx.x] = r;
    }
}

__global__ __launch_bounds__(kThreads)
void yolo_pass2(const BlockRec* __restrict__ recs, float* __restrict__ out) {
    const int tid = threadIdx.x;
    float s = 0.0f;
    P2 acc; acc.f0 = INT_MAX; acc.d0 = 0.0f; acc.f1 = INT_MAX; acc.d1 = 0.0f;
    for (int i = tid; i < kBlocks; i += kThreads) {
        BlockRec r = recs[i];
        s += r.sum;
        P2 q; q.f0 = r.f0; q.d0 = r.d0; q.f1 = r.f1; q.d1 = r.d1;
        acc = merge_p2(acc, q);
    }
    float ws = wave_reduce_sum_wmma(s);
    P2    wp = wave_reduce_p2(acc);

    __shared__ float swsum[kThreads / 32];
    __shared__ P2    swp2[kThreads / 32];
    const int wid = tid >> 5, lane = tid & 31;
    if (lane == 0) { swsum[wid] = ws; swp2[wid] = wp; }
    __syncthreads();
    if (tid == 0) {
        float tot = 0.0f;
        P2 a = swp2[0];
#pragma unroll
        for (int i = 0; i < kThreads / 32; ++i) tot += swsum[i];
#pragma unroll
        for (int i = 1; i < kThreads / 32; ++i) a = merge_p2(a, swp2[i]);
        float corr = 0.0f;                 // first two responsible boxes: sq -> plain
        if (a.f0 != INT_MAX) corr += a.d0;
        if (a.f1 != INT_MAX) corr += a.d1;
        out[0] = (tot + kLCoord * corr) * (1.0f / (float)kBatch);
    }
}

} // namespace

extern "C" void kernel_launch(void* const* d_in, const int* in_sizes, int n_in,
                              void* d_out, int out_size, void* d_ws, size_t ws_size,
                              hipStream_t stream) {
    const float* pred = (const float*)d_in[0];
    const float* targ = (const float*)d_in[1];
    BlockRec* recs = (BlockRec*)d_ws;       // 3136 * 24 B = ~75 KB scratch
    (void)in_sizes; (void)n_in; (void)out_size; (void)ws_size;

    yolo_pass1<<<kBlocks, kThreads, 0, stream>>>(pred, targ, recs);
    yolo_pass2<<<1, kThreads, 0, stream>>>(recs, (float*)d_out);
}